// Qfilter_28724741276152
// MI455X (gfx1250) — compile-verified
//
#include <hip/hip_runtime.h>

// MI455X / gfx1250 — bandwidth-bound elementwise quantum-filter kernel.
// out[i,0,j] = f(x[i,j], x[i,j+1], w) ; no contraction dim => no WMMA.
// Optimize the HBM stream: b128 NT loads, register-resolved a/b overlap,
// shared transcendentals (10 trig per 4 outputs), branchless halo.

typedef float v4f __attribute__((ext_vector_type(4)));

#define TLEN 4096   // inner length of x rows
#define OUTW 4095   // outputs per row (dilation 1)

__device__ __forceinline__ float qf_elem(float ca, float sa, float cb, float sb,
                                         float c0, float s0, float c1, float s1) {
    // psi after RY(a) wire0, RY(b) wire1 from (.5,.5,.5,.5)
    const float A   = 0.5f * (ca - sa);
    const float B   = 0.5f * (ca + sa);
    const float cmb = cb - sb;
    const float cpb = cb + sb;
    const float q0 = A * cmb;
    const float q1 = A * cpb;
    const float q2 = B * cmb;
    const float q3 = B * cpb;
    // CNOT01 swaps components 2<->3, then RY(w0) on wire0:
    const float t0 = c0 * q0 - s0 * q3;
    const float t1 = c0 * q1 - s0 * q2;
    const float t2 = s0 * q0 + c0 * q3;
    const float t3 = s0 * q1 + c0 * q2;
    // RY(w1) on wire1; only components 0 and 3 reach the output:
    const float u0 = c1 * t0 - s1 * t1;
    const float u3 = s1 * t2 + c1 * t3;
    return 2.0f * (u0 * u0 - u3 * u3);
}

__global__ __launch_bounds__(256) void qfilter_kernel(
    const float* __restrict__ x,      // (rows, 4096)
    const float* __restrict__ w,      // (1, 2)
    float* __restrict__ out)          // (rows, 1, 4095)
{
    // Wave-uniform weight rotations (scalar-cached loads).
    const float h0 = 0.5f * w[0];
    const float h1 = 0.5f * w[1];
    const float c0 = __cosf(h0), s0 = __sinf(h0);
    const float c1 = __cosf(h1), s1 = __sinf(h1);

    // 4 chunk-blocks of 1024 outputs per row; thread owns 4 consecutive outputs.
    const int bx  = blockIdx.x;
    const int row = bx >> 2;
    const int j0  = ((bx & 3) << 10) + ((int)threadIdx.x << 2);  // 0..4092, 16B aligned
    const float* xrow = x + (size_t)row * TLEN;

    // Read-once stream: NT 128-bit load + branchless clamped halo scalar.
    // (For j0==4092 the halo feeds only the culled j==4095 slot.)
    const v4f v = __builtin_nontemporal_load((const v4f*)(xrow + j0));
    const int jh = (j0 + 4 < TLEN) ? (j0 + 4) : (TLEN - 1);
    const float e4 = __builtin_nontemporal_load(xrow + jh);

    const float e[5] = { v.x, v.y, v.z, v.w, e4 };

    // Adjacent outputs share trig operands: 10 sin/cos per 4 outputs.
    float cc[5], ss[5];
#pragma unroll
    for (int k = 0; k < 5; ++k) {
        const float h = 0.5f * e[k];
        cc[k] = __cosf(h);
        ss[k] = __sinf(h);
    }

    float* orow = out + (size_t)row * OUTW + j0;
    // j0 <= 4092, so outputs j0..j0+2 are always in range; only k==3 can fall
    // on the nonexistent j==4095 slot.
#pragma unroll
    for (int k = 0; k < 3; ++k) {
        const float r = qf_elem(cc[k], ss[k], cc[k + 1], ss[k + 1], c0, s0, c1, s1);
        __builtin_nontemporal_store(r, orow + k);
    }
    if (j0 + 3 < OUTW) {
        const float r = qf_elem(cc[3], ss[3], cc[4], ss[4], c0, s0, c1, s1);
        __builtin_nontemporal_store(r, orow + 3);
    }
}

extern "C" void kernel_launch(void* const* d_in, const int* in_sizes, int n_in,
                              void* d_out, int out_size, void* d_ws, size_t ws_size,
                              hipStream_t stream) {
    const float* x   = (const float*)d_in[0];   // (1024, 4096) f32
    const float* wts = (const float*)d_in[1];   // (1, 2) f32
    float* out       = (float*)d_out;           // (1024, 1, 4095) f32

    const int rows = in_sizes[0] / TLEN;        // 1024
    dim3 grid(rows * 4);                        // 4 blocks of 256 threads per row
    dim3 block(256);                            // 8 wave32 waves per block
    qfilter_kernel<<<grid, block, 0, stream>>>(x, wts, out);
}